// CLSProcess_67723044323857
// MI455X (gfx1250) — compile-verified
//
#include <hip/hip_runtime.h>

// CDNA5 / gfx1250 chunked linear-recurrence scan, two-kernel design.
//
//   out[b,t,u] = y[b,t]*out[b,t-1,u] + x[b,t]*z[b,t,u]
//
// Kernel 1 (tiny): per (b, 16-step chunk) precompute, in WMMA A-fragment layout,
//   slots 0..3 : A'[i][j] = L[i][j]*x[j]   (L = lower-triangular decay products)
//   slot  4    : A5[i][0] = p_i = prod_{r<=i} y[r]   (carry coefficients)
// Kernel 2 (streaming): per (b, 16-unit tile), for each chunk do
//   acc = WMMA(A5, [state;0])  +  sum_kk WMMA(A'_kk, z_kk)     (5 x 16x16x4 f32)
// One bpermute per chunk extracts the new state (row 15).

typedef __attribute__((ext_vector_type(2))) float v2f;
typedef __attribute__((ext_vector_type(4))) float v4f;
typedef __attribute__((ext_vector_type(8))) float v8f;

#define B_     8
#define T_     4096
#define U_     1024
#define F_     (U_ + 2)
#define CH     16
#define NCH    (T_ / CH)
#define ALANE  12            // floats per lane per chunk in Apacked (48B, keeps b128 aligned)
#define ACHUNK (32 * ALANE)  // floats per (b,chunk) = 1536B; total 8*256*1536 = 3MB in d_ws

// ---------------------------------------------------------------- kernel 1 ----
__global__ __launch_bounds__(256)
void CLS_precompute_A(const float* __restrict__ in, float* __restrict__ Ap) {
  const int lane = threadIdx.x & 31;
  const int wid  = (int)((blockIdx.x * blockDim.x + threadIdx.x) >> 5);
  const int b    = wid >> 8;      // NCH = 256 chunks per batch
  const int c    = wid & (NCH - 1);
  const int t0   = c * CH;
  const int m    = lane & 15;
  const int hi   = lane >> 4;
  const int koff = hi << 1;

  const float2 v = *(const float2*)(in + (size_t)(b * T_ + t0 + m) * F_);
  const float xl = v.x;
  float pm = v.y;                         // inclusive product scan of y over the chunk
#pragma unroll
  for (int o = 1; o < 16; o <<= 1) {
    const float t = __shfl_up(pm, o, 16);
    pm = (m >= o) ? pm * t : pm;
  }
  const float rq = 1.0f / pm;             // lane m holds 1/p_m

  float av[10];
#pragma unroll
  for (int kk = 0; kk < 4; ++kk) {
    const int c0 = kk * 4 + koff;
    const float inv0 = __shfl(rq, c0, 32);
    const float inv1 = __shfl(rq, c0 + 1, 32);
    const float x0   = __shfl(xl, c0, 32);
    const float x1   = __shfl(xl, c0 + 1, 32);
    av[kk * 2 + 0] = (c0     <= m) ? pm * inv0 * x0 : 0.0f;  // L[m][c]*x[c]
    av[kk * 2 + 1] = (c0 + 1 <= m) ? pm * inv1 * x1 : 0.0f;
  }
  av[8] = hi ? 0.0f : pm;                 // carry slot: A5[m][0] = p_m (K=0 only)
  av[9] = 0.0f;

  float* dst = Ap + (size_t)(b * NCH + c) * ACHUNK + lane * ALANE;
  v4f w0 = {av[0], av[1], av[2], av[3]};
  v4f w1 = {av[4], av[5], av[6], av[7]};
  v2f w2 = {av[8], av[9]};
  *(v4f*)(dst + 0) = w0;
  *(v4f*)(dst + 4) = w1;
  *(v2f*)(dst + 8) = w2;
}

// ---------------------------------------------------------------- kernel 2 ----
__global__ __launch_bounds__(256)
void CLS_scan_wmma_main(const float* __restrict__ in,
                        const float* __restrict__ Ap,
                        float* __restrict__ out) {
  const int lane = threadIdx.x & 31;
  const int wid  = (int)((blockIdx.x * blockDim.x + threadIdx.x) >> 5);
  const int b    = wid >> 6;          // 64 unit-tiles per batch
  const int u0   = (wid & 63) << 4;   // 16 units per tile
  const int m    = lane & 15;
  const int hi   = lane >> 4;
  const int koff = hi << 1;

  // ---- software pipeline registers: A fragment + z tile of current chunk ----
  v4f a01, a23; v2f a4; float zb[8];
  {
    const float* ab = Ap + (size_t)(b * NCH + 0) * ACHUNK + lane * ALANE;
    a01 = *(const v4f*)(ab + 0);
    a23 = *(const v4f*)(ab + 4);
    a4  = *(const v2f*)(ab + 8);
#pragma unroll
    for (int kk = 0; kk < 4; ++kk) {
      const int tb = kk * 4 + koff;
#pragma unroll
      for (int j = 0; j < 2; ++j)
        zb[kk * 2 + j] = in[(size_t)(b * T_ + tb + j) * F_ + 2 + u0 + m];
    }
  }

  float st = 0.0f;  // state for unit u0+m (zero initial condition)

  for (int c = 0; c < NCH; ++c) {
    const int t0 = c * CH;

    // ---- prefetch next chunk (independent of the scan dependence) ----
    v4f a01n = {}, a23n = {}; v2f a4n = {}; float zn[8];
#pragma unroll
    for (int j = 0; j < 8; ++j) zn[j] = 0.0f;
    if (c + 1 < NCH) {
      const float* ab = Ap + (size_t)(b * NCH + c + 1) * ACHUNK + lane * ALANE;
      a01n = *(const v4f*)(ab + 0);
      a23n = *(const v4f*)(ab + 4);
      a4n  = *(const v2f*)(ab + 8);
      const int t1 = t0 + CH;
#pragma unroll
      for (int kk = 0; kk < 4; ++kk) {
        const int tb = t1 + kk * 4 + koff;
#pragma unroll
        for (int j = 0; j < 2; ++j)
          zn[kk * 2 + j] = in[(size_t)(b * T_ + tb + j) * F_ + 2 + u0 + m];
      }
    }
    if (c + 2 < NCH)  // L2 hint two chunks ahead (global_prefetch_b8)
      __builtin_prefetch(in + (size_t)(b * T_ + t0 + 2 * CH + m) * F_ + 2 + u0, 0, 1);

    // ---- carry as rank-1 WMMA: acc = A5 x [state; 0] ----
    v2f b5;
    b5.x = hi ? 0.0f : st;   // B row K=0 = state vector, all other rows 0
    b5.y = 0.0f;
    v8f acc = {};
    acc = __builtin_amdgcn_wmma_f32_16x16x4_f32(false, a4, false, b5,
                                                (short)0, acc, false, false);
    // ---- 4 K-slices of the triangular matmul ----
#pragma unroll
    for (int kk = 0; kk < 4; ++kk) {
      v2f a, bm;
      if (kk == 0)      { a.x = a01.x; a.y = a01.y; }
      else if (kk == 1) { a.x = a01.z; a.y = a01.w; }
      else if (kk == 2) { a.x = a23.x; a.y = a23.y; }
      else              { a.x = a23.z; a.y = a23.w; }
      bm.x = zb[kk * 2 + 0];
      bm.y = zb[kk * 2 + 1];
      acc = __builtin_amdgcn_wmma_f32_16x16x4_f32(false, a, false, bm,
                                                  (short)0, acc, false, false);
    }

    // ---- new state = row 15 of the tile (lanes 16-31, vgpr 7) ----
    st = __shfl(acc[7], m + 16, 32);

    // ---- store 16(time) x 16(unit) tile: 64B contiguous per half-wave ----
#pragma unroll
    for (int v = 0; v < 8; ++v) {
      const int t = t0 + v + (hi << 3);
      out[(size_t)(b * T_ + t) * U_ + u0 + m] = acc[v];
    }

    // rotate pipeline
    a01 = a01n; a23 = a23n; a4 = a4n;
#pragma unroll
    for (int j = 0; j < 8; ++j) zb[j] = zn[j];
  }
}

// ---------------------------------------------------------------- launcher ----
extern "C" void kernel_launch(void* const* d_in, const int* in_sizes, int n_in,
                              void* d_out, int out_size, void* d_ws, size_t ws_size,
                              hipStream_t stream) {
  (void)in_sizes; (void)n_in; (void)out_size; (void)ws_size;
  const float* in = (const float*)d_in[0];
  float* out = (float*)d_out;
  float* Ap  = (float*)d_ws;  // 3 MB of scratch for packed A fragments

  // Kernel 1: 2048 waves (8 batches x 256 chunks), 8 waves per block.
  CLS_precompute_A<<<dim3(B_ * NCH * 32 / 256), dim3(256), 0, stream>>>(in, Ap);
  // Kernel 2: 512 waves (8 batches x 64 unit-tiles), 8 waves per block.
  CLS_scan_wmma_main<<<dim3(B_ * (U_ / 16) * 32 / 256), dim3(256), 0, stream>>>(in, Ap, out);
}